// GraphSAGE_10694468567288
// MI455X (gfx1250) — compile-verified
//
#include <hip/hip_runtime.h>
#include <math.h>

typedef __attribute__((ext_vector_type(2))) float v2f;
typedef __attribute__((ext_vector_type(8))) float v8f;

#define NNODES 50000
#define NEDGES 800000
#define KMAX   256
#define KPADMX (KMAX + 4)   // +4 floats: lane stride 260 % 64 banks = 4 -> conflict-free

// ---------------- utility kernels ----------------

__global__ void zero_f32(float* __restrict__ p, int n) {
    int i = blockIdx.x * blockDim.x + threadIdx.x;
    if (i < n) p[i] = 0.0f;
}

__global__ void degree_kernel(const int* __restrict__ dst, float* __restrict__ deg, int nE) {
    int e = blockIdx.x * blockDim.x + threadIdx.x;
    if (e < nE) atomicAdd(&deg[dst[e]], 1.0f);
}

// One block per edge; blockDim.x == feature width F. Row-contiguous gather +
// row-contiguous atomic scatter -> coalesced full-row transactions; the whole
// activation set (~130MB) is resident in the MI455X 192MB L2, so these random
// row gathers/atomics resolve at L2 speed, not HBM.
__global__ void scatter_add_kernel(const float* __restrict__ x,
                                   const int* __restrict__ src,
                                   const int* __restrict__ dst,
                                   float* __restrict__ agg) {
    const int e = blockIdx.x;
    const int f = threadIdx.x;
    const int F = blockDim.x;
    const int s = src[e];
    const int d = dst[e];
    atomicAdd(&agg[(size_t)d * F + f], x[(size_t)s * F + f]);
}

// One block per node row; deg fetched once, no per-element integer division.
__global__ void mean_div_kernel(float* __restrict__ agg, const float* __restrict__ deg) {
    const int node = blockIdx.x;
    const int F = blockDim.x;
    const float inv = 1.0f / fmaxf(deg[node], 1.0f);
    agg[(size_t)node * F + threadIdx.x] *= inv;
}

// ---------------- fused SAGE layer: out = [relu](Amean @ Wl + b + Ax @ Wr) ----
// grid = (ceil(rowTiles/4), M/16); 4 waves/block, one 16x16 tile per wave.
// Block stages its 16-column slice of Wl and Wr into LDS (transposed + padded),
// then every WMMA k-step reads B as one conflict-free ds_load_b64 and A as one
// aligned global_load_b64. V_WMMA_F32_16X16X4_F32 keeps full fp32 semantics.
__global__ void __launch_bounds__(128)
sage_gemm_wmma(const float* __restrict__ Amean,
               const float* __restrict__ Wl,
               const float* __restrict__ Ax,
               const float* __restrict__ Wr,
               const float* __restrict__ bias,
               float* __restrict__ out,
               int Nrows, int K, int M, int do_relu) {
    __shared__ float ldsW[2][16 * KPADMX];

    const int lane = threadIdx.x & 31;
    const int wave = threadIdx.x >> 5;
    const int Kpad = K + 4;
    const int col0 = blockIdx.y << 4;

    // ---- cooperative stage of the 16-col weight slices, transposed ----
    // ldsW[m][c * Kpad + k] = W[k * M + col0 + c]
    for (int idx = threadIdx.x; idx < 16 * K; idx += blockDim.x) {
        const int k = idx >> 4;
        const int c = idx & 15;
        ldsW[0][c * Kpad + k] = Wl[(size_t)k * M + col0 + c];
        ldsW[1][c * Kpad + k] = Wr[(size_t)k * M + col0 + c];
    }
    __syncthreads();

    const int rowTile   = blockIdx.x * 4 + wave;       // wave-uniform
    const int nRowTiles = Nrows >> 4;
    if (rowTile < nRowTiles) {                         // wave-uniform guard: EXEC all-1s
        const int row0 = rowTile << 4;
        const int half = lane >> 4;                    // selects K pair {k,k+1} / {k+2,k+3}
        const int l15  = lane & 15;
        const int arow = row0 + l15;
        const int bcol = col0 + l15;

        const float* __restrict__ Arow  = Amean + (size_t)arow * K;
        const float* __restrict__ Xrow  = Ax    + (size_t)arow * K;
        const float* __restrict__ ldsBl = &ldsW[0][l15 * Kpad];
        const float* __restrict__ ldsBr = &ldsW[1][l15 * Kpad];

        v8f acc = {};

        // pass 1: mean @ Wl
        for (int k = 0; k < K; k += 4) {
            const int ka = k + (half << 1);            // even -> 8B-aligned b64 loads
            const v2f a = *(const v2f*)(Arow + ka);
            const v2f b = *(const v2f*)(ldsBl + ka);
            acc = __builtin_amdgcn_wmma_f32_16x16x4_f32(false, a, false, b,
                                                        (short)0, acc, false, false);
        }
        // pass 2: x @ Wr (same accumulator)
        for (int k = 0; k < K; k += 4) {
            const int ka = k + (half << 1);
            const v2f a = *(const v2f*)(Xrow + ka);
            const v2f b = *(const v2f*)(ldsBr + ka);
            acc = __builtin_amdgcn_wmma_f32_16x16x4_f32(false, a, false, b,
                                                        (short)0, acc, false, false);
        }

        // C/D layout: VGPR v -> row M=v (lanes 0-15) / M=v+8 (lanes 16-31), col = lane&15
        const float bv = bias[bcol];
        #pragma unroll
        for (int v = 0; v < 8; ++v) {
            const int r = row0 + v + (half << 3);
            float val = acc[v] + bv;
            if (do_relu) val = fmaxf(val, 0.0f);
            out[(size_t)r * M + bcol] = val;
        }
    }
}

// ---------------- row-wise log_softmax over 64 columns (wave32: 2 cols/lane) ---
__global__ void logsoftmax64_kernel(const float* __restrict__ h, float* __restrict__ out,
                                    int Nrows) {
    const int row = blockIdx.x * (blockDim.x >> 5) + (threadIdx.x >> 5);
    if (row >= Nrows) return;
    const int lane = threadIdx.x & 31;
    const float v0 = h[(size_t)row * 64 + lane];
    const float v1 = h[(size_t)row * 64 + lane + 32];

    float m = fmaxf(v0, v1);
    #pragma unroll
    for (int off = 16; off > 0; off >>= 1) m = fmaxf(m, __shfl_xor(m, off, 32));

    float s = expf(v0 - m) + expf(v1 - m);
    #pragma unroll
    for (int off = 16; off > 0; off >>= 1) s += __shfl_xor(s, off, 32);

    const float lse = m + logf(s);
    out[(size_t)row * 64 + lane]      = v0 - lse;
    out[(size_t)row * 64 + lane + 32] = v1 - lse;
}

// ---------------- launcher ----------------

extern "C" void kernel_launch(void* const* d_in, const int* in_sizes, int n_in,
                              void* d_out, int out_size, void* d_ws, size_t ws_size,
                              hipStream_t stream) {
    const float* x   = (const float*)d_in[0];
    const int*   ei  = (const int*)d_in[1];
    const int*   src = ei;
    const int*   dst = ei + NEDGES;
    const float* W1l = (const float*)d_in[2];
    const float* b1  = (const float*)d_in[3];
    const float* W1r = (const float*)d_in[4];
    const float* W2l = (const float*)d_in[5];
    const float* b2  = (const float*)d_in[6];
    const float* W2r = (const float*)d_in[7];
    const float* W3l = (const float*)d_in[8];
    const float* b3  = (const float*)d_in[9];
    const float* W3r = (const float*)d_in[10];

    const int N = NNODES, E = NEDGES;
    const int rowTiles  = N / 16;                       // 3125
    const int gemmGridX = (rowTiles + 3) / 4;           // 782 blocks of 4 waves

    float* ws  = (float*)d_ws;
    float* deg = ws;                                    // N
    float* buf = deg + N;                               // N*256 scratch (agg/mean)
    float* h1  = buf + (size_t)N * 256;                 // N*256
    float* h2  = h1 + (size_t)N * 256;                  // N*128
    float* out_h  = (float*)d_out;                      // N*64
    float* out_ls = out_h + (size_t)N * 64;             // N*64

    // degree (shared by all three layers)
    zero_f32<<<(N + 255) / 256, 256, 0, stream>>>(deg, N);
    degree_kernel<<<(E + 255) / 256, 256, 0, stream>>>(dst, deg, E);

    // ---- layer 1: 256 -> 256, relu ----
    zero_f32<<<(N * 256 + 255) / 256, 256, 0, stream>>>(buf, N * 256);
    scatter_add_kernel<<<E, 256, 0, stream>>>(x, src, dst, buf);
    mean_div_kernel<<<N, 256, 0, stream>>>(buf, deg);
    sage_gemm_wmma<<<dim3(gemmGridX, 256 / 16), 128, 0, stream>>>(
        buf, W1l, x, W1r, b1, h1, N, 256, 256, 1);

    // ---- layer 2: 256 -> 128, relu ----
    zero_f32<<<(N * 256 + 255) / 256, 256, 0, stream>>>(buf, N * 256);
    scatter_add_kernel<<<E, 256, 0, stream>>>(h1, src, dst, buf);
    mean_div_kernel<<<N, 256, 0, stream>>>(buf, deg);
    sage_gemm_wmma<<<dim3(gemmGridX, 128 / 16), 128, 0, stream>>>(
        buf, W2l, h1, W2r, b2, h2, N, 256, 128, 1);

    // ---- layer 3: 128 -> 64, no relu, writes directly into d_out ----
    zero_f32<<<(N * 128 + 255) / 256, 256, 0, stream>>>(buf, N * 128);
    scatter_add_kernel<<<E, 128, 0, stream>>>(h2, src, dst, buf);
    mean_div_kernel<<<N, 128, 0, stream>>>(buf, deg);
    sage_gemm_wmma<<<dim3(gemmGridX, 64 / 16), 128, 0, stream>>>(
        buf, W3l, h2, W3r, b3, out_h, N, 128, 64, 0);

    // ---- log_softmax over 64 classes ----
    logsoftmax64_kernel<<<(N + 7) / 8, 256, 0, stream>>>(out_h, out_ls, N);
}